// GCNEncoder_6597069767366
// MI455X (gfx1250) — compile-verified
//
#include <hip/hip_runtime.h>

// ---------------------------------------------------------------------------
// GCN encoder for MI455X (gfx1250, wave32, WMMA).
// f32 end-to-end: GEMMs use V_WMMA_F32_16X16X4_F32 (exact f32 matrix math).
// W is staged TRANSPOSED+PADDED in LDS so each B fragment is one aligned
// ds_load_b64 straight into the WMMA operand pair (no repack movs, and the
// 32 lanes' b64 reads tile all 64 LDS banks exactly once).
// SpMM: one wave32 per edge, float4 gathers, hw f32 global atomics (the
// 25.6MB support matrix is resident in the 192MB L2).
// ---------------------------------------------------------------------------

typedef __attribute__((ext_vector_type(2))) float v2f;
typedef __attribute__((ext_vector_type(8))) float v8f;

#define N_NODES 50000
#define EMB 128
#define HID 128
#define HALF 64

// ---------------------------------------------------------------------------
// WMMA f32 GEMM:  Out[nrows x HOUT] = A[nrows x K] @ W[K x HOUT] (+bias)(ReLU)
// Block: 256 threads = 8 waves; each wave owns a 16-row strip, covers all
// HOUT/16 column tiles. W staged transposed in LDS with stride K+4 dwords:
// bank(dword) = (4*n + k) mod 64 -> conflict-free b64 loads across the wave.
// ---------------------------------------------------------------------------
template <int K, int HOUT, bool BIAS, bool RELU>
__global__ __launch_bounds__(256) void gemm_wmma_f32(
    const float* __restrict__ A, const float* __restrict__ W,
    const float* __restrict__ bias, float* __restrict__ Out, int nrows) {
  constexpr int LDK = K + 4;  // padded stride (even: keeps b64 loads 8B-aligned)
  __shared__ float wlds[HOUT * LDK];
  for (int i = threadIdx.x; i < K * HOUT; i += 256) {
    const int k = i / HOUT;
    const int n = i - k * HOUT;
    wlds[n * LDK + k] = W[i];  // transpose: column n contiguous in k
  }
  __syncthreads();

  const int lane = threadIdx.x & 31;
  const int wave = threadIdx.x >> 5;
  const int rowBase = blockIdx.x * 128 + wave * 16;
  if (rowBase >= nrows) return;  // whole wave out of range

  const int half = lane >> 4;  // 0: lanes 0-15, 1: lanes 16-31
  const int l15 = lane & 15;

  constexpr int NT = HOUT / 16;
  v8f acc[NT];
  const v8f vzero = {0.f, 0.f, 0.f, 0.f, 0.f, 0.f, 0.f, 0.f};
#pragma unroll
  for (int t = 0; t < NT; ++t) acc[t] = vzero;

  // A fragment rows: lanes 0-15 -> M = rowBase + l15 (clamped; stores guarded)
  const int m = rowBase + l15;
  const float* arow = A + (size_t)((m < nrows) ? m : (nrows - 1)) * K;

  // Per-lane base into the transposed W tile for this lane's column(s).
  const float* bcol[NT];
#pragma unroll
  for (int t = 0; t < NT; ++t) bcol[t] = &wlds[(t * 16 + l15) * LDK + 2 * half];

  for (int k0 = 0; k0 < K; k0 += 4) {
    // 32-bit A 16x4 layout: V0 = K{0,2}, V1 = K{1,3} across lane halves.
    v2f a = *(const v2f*)(arow + k0 + 2 * half);
#pragma unroll
    for (int t = 0; t < NT; ++t) {
      v2f b = *(const v2f*)(bcol[t] + k0);  // one ds_load_b64, pair-aligned
      acc[t] = __builtin_amdgcn_wmma_f32_16x16x4_f32(
          /*neg_a=*/false, a, /*neg_b=*/false, b,
          /*c_mod=*/(short)0, acc[t], /*reuse_a=*/false, /*reuse_b=*/false);
    }
  }

  // C/D layout: VGPR r -> rows (rowBase+r+8*half); cols = t*16 + l15.
  if (rowBase + 16 <= nrows) {  // fast path: full tile, no guards
#pragma unroll
    for (int t = 0; t < NT; ++t) {
      const int n = t * 16 + l15;
      const float bv = BIAS ? bias[n] : 0.0f;
#pragma unroll
      for (int r = 0; r < 8; ++r) {
        float v = acc[t][r] + bv;
        if (RELU) v = fmaxf(v, 0.0f);
        Out[(size_t)(rowBase + r + half * 8) * HOUT + n] = v;
      }
    }
  } else {
#pragma unroll
    for (int t = 0; t < NT; ++t) {
      const int n = t * 16 + l15;
      const float bv = BIAS ? bias[n] : 0.0f;
#pragma unroll
      for (int r = 0; r < 8; ++r) {
        const int mm = rowBase + r + half * 8;
        if (mm < nrows) {
          float v = acc[t][r] + bv;
          if (RELU) v = fmaxf(v, 0.0f);
          Out[(size_t)mm * HOUT + n] = v;
        }
      }
    }
  }
}

// ---------------------------------------------------------------------------
// SpMM scatter: agg[row[e], :] += val[e] * S[col[e], :]
// One wave32 per edge: 32 lanes x float4 = 128 channels.
// ---------------------------------------------------------------------------
__global__ __launch_bounds__(256) void spmm_edge_atomic(
    const int* __restrict__ erow, const int* __restrict__ ecol,
    const float* __restrict__ eval, const float* __restrict__ S,
    float* __restrict__ agg, int nedges) {
  const int e = blockIdx.x * 8 + (threadIdx.x >> 5);
  if (e >= nedges) return;
  const int lane = threadIdx.x & 31;
  const int r = erow[e];
  const int c = ecol[e];
  const float v = eval[e];
  const float4 s = ((const float4*)(S + (size_t)c * HID))[lane];
  float* ap = agg + (size_t)r * HID + lane * 4;
  unsafeAtomicAdd(ap + 0, v * s.x);
  unsafeAtomicAdd(ap + 1, v * s.y);
  unsafeAtomicAdd(ap + 2, v * s.z);
  unsafeAtomicAdd(ap + 3, v * s.w);
}

// ---------------------------------------------------------------------------
// Utility: zero buffer (float4 stores), bias+ReLU (in-place safe)
// ---------------------------------------------------------------------------
__global__ __launch_bounds__(256) void zero_f32(float4* __restrict__ p,
                                                int n4) {
  int i = blockIdx.x * 256 + threadIdx.x;
  if (i < n4) p[i] = make_float4(0.f, 0.f, 0.f, 0.f);
}

__global__ __launch_bounds__(256) void bias_relu_128(
    const float* __restrict__ in, const float* __restrict__ bias,
    float* __restrict__ out, int nrows) {
  int t = blockIdx.x * 256 + threadIdx.x;
  int total = nrows * (HID / 4);
  if (t >= total) return;
  int c4 = t & 31;  // 32 float4 per row
  int row = t >> 5;
  float4 x = ((const float4*)(in + (size_t)row * HID))[c4];
  float4 b = ((const float4*)bias)[c4];
  float4 y;
  y.x = fmaxf(x.x + b.x, 0.f);
  y.y = fmaxf(x.y + b.y, 0.f);
  y.z = fmaxf(x.z + b.z, 0.f);
  y.w = fmaxf(x.w + b.w, 0.f);
  ((float4*)(out + (size_t)row * HID))[c4] = y;
}

// ---------------------------------------------------------------------------
// Orchestration
// ---------------------------------------------------------------------------
extern "C" void kernel_launch(void* const* d_in, const int* in_sizes, int n_in,
                              void* d_out, int out_size, void* d_ws,
                              size_t ws_size, hipStream_t stream) {
  const float* x = (const float*)d_in[0];
  const int* erow = (const int*)d_in[1];
  const int* ecol = (const int*)d_in[2];
  const float* eval = (const float*)d_in[3];
  const float* W_gc0 = (const float*)d_in[4];
  const float* b_gc0 = (const float*)d_in[5];
  const float* W_gc1 = (const float*)d_in[6];
  const float* b_gc1 = (const float*)d_in[7];
  const float* Wm1 = (const float*)d_in[8];
  const float* bm1 = (const float*)d_in[9];
  const float* Wm2 = (const float*)d_in[10];
  const float* bm2 = (const float*)d_in[11];
  const float* Wv1 = (const float*)d_in[12];
  const float* bv1 = (const float*)d_in[13];
  const float* Wv2 = (const float*)d_in[14];
  const float* bv2 = (const float*)d_in[15];

  const int N = N_NODES;
  const int E = in_sizes[1];

  float* ws0 = (float*)d_ws;           // N*128
  float* ws1 = ws0 + (size_t)N * HID;  // N*128
  float* ws2 = ws1 + (size_t)N * HID;  // N*128

  float* out_mean = (float*)d_out;
  float* out_logvar = out_mean + (size_t)N * HALF;

  const int gemmGrid = (N + 127) / 128;
  const int zeroGrid = ((N * HID / 4) + 255) / 256;
  const int brGrid = ((N * (HID / 4)) + 255) / 256;
  const int spmmGrid = (E + 7) / 8;

  // ---- GCN layer 0: support = x @ W_gc0 ; agg = A*support ; h = relu(agg+b)
  gemm_wmma_f32<EMB, HID, false, false>
      <<<gemmGrid, 256, 0, stream>>>(x, W_gc0, nullptr, ws0, N);
  zero_f32<<<zeroGrid, 256, 0, stream>>>((float4*)ws1, N * HID / 4);
  spmm_edge_atomic<<<spmmGrid, 256, 0, stream>>>(erow, ecol, eval, ws0, ws1, E);
  bias_relu_128<<<brGrid, 256, 0, stream>>>(ws1, b_gc0, ws1, N);

  // ---- GCN layer 1
  gemm_wmma_f32<HID, HID, false, false>
      <<<gemmGrid, 256, 0, stream>>>(ws1, W_gc1, nullptr, ws0, N);
  zero_f32<<<zeroGrid, 256, 0, stream>>>((float4*)ws2, N * HID / 4);
  spmm_edge_atomic<<<spmmGrid, 256, 0, stream>>>(erow, ecol, eval, ws0, ws2, E);
  bias_relu_128<<<brGrid, 256, 0, stream>>>(ws2, b_gc1, ws2, N);

  // ---- mean head: t = relu(h@Wm1+bm1) ; mean = t@Wm2+bm2
  gemm_wmma_f32<HID, HALF, true, true>
      <<<gemmGrid, 256, 0, stream>>>(ws2, Wm1, bm1, ws0, N);
  gemm_wmma_f32<HALF, HALF, true, false>
      <<<gemmGrid, 256, 0, stream>>>(ws0, Wm2, bm2, out_mean, N);

  // ---- logvar head
  gemm_wmma_f32<HID, HALF, true, true>
      <<<gemmGrid, 256, 0, stream>>>(ws2, Wv1, bv1, ws1, N);
  gemm_wmma_f32<HALF, HALF, true, false>
      <<<gemmGrid, 256, 0, stream>>>(ws1, Wv2, bv2, out_logvar, N);
}